// VGGSmall_77034533421673
// MI455X (gfx1250) — compile-verified
//
#include <hip/hip_runtime.h>

typedef __attribute__((ext_vector_type(2))) float v2f;
typedef __attribute__((ext_vector_type(8))) float v8f;

#define BN_EPS 1e-5f

// ---------------------------------------------------------------------------
// Fused AdderNet layer: out[n,co,h,w] = relu( bn( -sum_{ci,kh,kw} |x - w| ) )
// Each thread computes a horizontal PAIR of output pixels for COB output
// channels. The shared 3x4 input window (12 loads) feeds 2*COB*9*2 = 288
// VALU ops per input channel, and every scalar-loaded weight value feeds two
// abs-diff accumulations. Weight/BN indexing is wave-uniform -> SMEM loads,
// so the inner loop is pure VALU: v_sub_f32 + v_add_f32 with |src| modifier.
// ---------------------------------------------------------------------------
template <int CIN, int COB>
__global__ __launch_bounds__(256) void adder_bn_relu_kern(
    const float* __restrict__ x,   // [B, CIN, H, W]
    const float* __restrict__ w,   // [Co, CIN, 3, 3]
    const float* __restrict__ bn,  // [4, Co] = gamma, beta, mean, var
    float* __restrict__ y,         // [B, Co, H, W]
    int B, int Co, int H, int W) {
  const int HW = H * W;
  const int PW = W >> 1;  // pixel pairs per row (W is even at every stage)
  int idx = blockIdx.x * 256 + threadIdx.x;
  if (idx >= B * H * PW) return;
  int t = idx;
  int pw = t % PW; t /= PW;
  int h  = t % H;  t /= H;
  int n  = t;
  int w0 = pw * 2;  // left pixel of the pair
  int cog = blockIdx.y * COB;

  float acc0[COB], acc1[COB];
#pragma unroll
  for (int c = 0; c < COB; ++c) { acc0[c] = 0.f; acc1[c] = 0.f; }

  const float* xn = x + (size_t)n * CIN * HW;
  const float* wg = w + (size_t)cog * CIN * 9;

  for (int ci = 0; ci < CIN; ++ci) {
    const float* xc = xn + ci * HW;
    if (ci + 1 < CIN) __builtin_prefetch(xc + HW + h * W + w0, 0, 0);

    // 3x4 window (rows h-1..h+1, cols w0-1..w0+2) with zero padding
    float win[12];
#pragma unroll
    for (int kh = 0; kh < 3; ++kh) {
      int hh = h + kh - 1;
      bool hin = ((unsigned)hh < (unsigned)H);
#pragma unroll
      for (int c4 = 0; c4 < 4; ++c4) {
        int col = w0 - 1 + c4;
        bool in = hin && ((unsigned)col < (unsigned)W);
        win[kh * 4 + c4] = in ? xc[hh * W + col] : 0.f;
      }
    }

#pragma unroll
    for (int c = 0; c < COB; ++c) {
      const float* wp = wg + (c * CIN + ci) * 9;  // uniform -> s_load
#pragma unroll
      for (int kh = 0; kh < 3; ++kh) {
#pragma unroll
        for (int kw = 0; kw < 3; ++kw) {
          float wv = wp[kh * 3 + kw];
          acc0[c] += fabsf(win[kh * 4 + kw]     - wv);
          acc1[c] += fabsf(win[kh * 4 + kw + 1] - wv);
        }
      }
    }
  }

#pragma unroll
  for (int c = 0; c < COB; ++c) {
    int co = cog + c;
    float g  = bn[0 * Co + co];
    float bb = bn[1 * Co + co];
    float m  = bn[2 * Co + co];
    float v  = bn[3 * Co + co];
    float inv  = g * rsqrtf(v + BN_EPS);
    float bias = bb - m * inv;
    v2f o;
    o.x = fmaxf((-acc0[c]) * inv + bias, 0.f);
    o.y = fmaxf((-acc1[c]) * inv + bias, 0.f);
    *(v2f*)(y + ((size_t)n * Co + co) * HW + h * W + w0) = o;  // b64 store
  }
}

// ---------------------------------------------------------------------------
// 2x2 max pool, one thread per output element.
// ---------------------------------------------------------------------------
__global__ __launch_bounds__(256) void maxpool2_kern(
    const float* __restrict__ in, float* __restrict__ out,
    int C, int H, int W, int total) {
  int i = blockIdx.x * 256 + threadIdx.x;
  if (i >= total) return;
  int Wo = W >> 1, Ho = H >> 1;
  int wq = i % Wo;
  int t = i / Wo;
  int hq = t % Ho;
  t /= Ho;
  int c = t % C;
  int n = t / C;
  const float* pp = in + (((size_t)n * C + c) * H + 2 * hq) * W + 2 * wq;
  out[i] = fmaxf(fmaxf(pp[0], pp[1]), fmaxf(pp[W], pp[W + 1]));
}

// ---------------------------------------------------------------------------
// FC head: out[128,10] = X[128,2048] @ fc_w[10,2048]^T + fc_b  via fp32 WMMA.
// One wave per 16-row batch tile; N padded 10 -> 16 (OOB lanes read row 0 of
// fc_w, their output columns are simply not stored).
// A fragment (16x4 f32, 2 VGPR): lanes 0-15 hold rows M, K = k0,k0+1;
// lanes 16-31 hold K = k0+2,k0+3.  B fragment mirrored with N across lanes.
// ---------------------------------------------------------------------------
__global__ __launch_bounds__(32) void fc_wmma_kern(
    const float* __restrict__ X,     // [128, 2048]
    const float* __restrict__ Wt,    // [10, 2048]
    const float* __restrict__ bias,  // [10]
    float* __restrict__ out) {       // [128, 10]
  const int K = 2048, N = 10;
  int lane = threadIdx.x & 31;
  int rt = blockIdx.x;        // row tile 0..7
  int half = lane >> 4;       // 0 or 1  -> K sub-pair
  int l15 = lane & 15;        // M row (A) / N col (B)

  const float* arow = X + (size_t)(rt * 16 + l15) * K + half * 2;
  const float* brow = Wt + (size_t)(l15 < N ? l15 : 0) * K + half * 2;

  v8f c = {};
  for (int k0 = 0; k0 < K; k0 += 4) {
    v2f a = *(const v2f*)(arow + k0);
    v2f b = *(const v2f*)(brow + k0);
    // 8 args: (neg_a, A, neg_b, B, c_mod, C, reuse_a, reuse_b)
    c = __builtin_amdgcn_wmma_f32_16x16x4_f32(
        false, a, false, b, (short)0, c, false, false);
  }

  if (l15 < N) {
    float bv = bias[l15];
#pragma unroll
    for (int j = 0; j < 8; ++j) {
      int m = rt * 16 + half * 8 + j;  // C/D layout: vgpr j -> M=j (lo), M=8+j (hi)
      out[m * N + l15] = c[j] + bv;
    }
  }
}

// ---------------------------------------------------------------------------
// Launch: ping-pong two 16MB workspace buffers through the whole network.
// ---------------------------------------------------------------------------
extern "C" void kernel_launch(void* const* d_in, const int* in_sizes, int n_in,
                              void* d_out, int out_size, void* d_ws, size_t ws_size,
                              hipStream_t stream) {
  (void)in_sizes; (void)n_in; (void)out_size; (void)ws_size;

  const float* x    = (const float*)d_in[0];
  const float* w1a  = (const float*)d_in[1];
  const float* bn1a = (const float*)d_in[2];
  const float* w1b  = (const float*)d_in[3];
  const float* bn1b = (const float*)d_in[4];
  const float* w2a  = (const float*)d_in[5];
  const float* bn2a = (const float*)d_in[6];
  const float* w2b  = (const float*)d_in[7];
  const float* bn2b = (const float*)d_in[8];
  const float* w3a  = (const float*)d_in[9];
  const float* bn3a = (const float*)d_in[10];
  const float* w3b  = (const float*)d_in[11];
  const float* bn3b = (const float*)d_in[12];
  const float* w3c  = (const float*)d_in[13];
  const float* bn3c = (const float*)d_in[14];
  const float* w3d  = (const float*)d_in[15];
  const float* bn3d = (const float*)d_in[16];
  const float* fc_w = (const float*)d_in[17];
  const float* fc_b = (const float*)d_in[18];
  float* out = (float*)d_out;

  float* A = (float*)d_ws;                                   // 16 MB
  float* Bb = (float*)((char*)d_ws + (size_t)(16u << 20));   // 16 MB

  dim3 blk(256);

  // Pixel-pair thread counts: B*H*(W/2)
  // Stage 1: 32x32 -> 128*32*16 = 65536 -> 256 blocks
  adder_bn_relu_kern<3, 8><<<dim3(256, 4), blk, 0, stream>>>(x, w1a, bn1a, A, 128, 32, 32, 32);
  adder_bn_relu_kern<32, 8><<<dim3(256, 4), blk, 0, stream>>>(A, w1b, bn1b, Bb, 128, 32, 32, 32);
  maxpool2_kern<<<dim3((128 * 32 * 16 * 16 + 255) / 256), blk, 0, stream>>>(Bb, A, 32, 32, 32, 128 * 32 * 16 * 16);

  // Stage 2: 16x16 -> 128*16*8 = 16384 -> 64 blocks
  adder_bn_relu_kern<32, 8><<<dim3(64, 8), blk, 0, stream>>>(A, w2a, bn2a, Bb, 128, 64, 16, 16);
  adder_bn_relu_kern<64, 8><<<dim3(64, 8), blk, 0, stream>>>(Bb, w2b, bn2b, A, 128, 64, 16, 16);
  maxpool2_kern<<<dim3((128 * 64 * 8 * 8 + 255) / 256), blk, 0, stream>>>(A, Bb, 64, 16, 16, 128 * 64 * 8 * 8);

  // Stage 3: 8x8 -> 128*8*4 = 4096 -> 16 blocks
  adder_bn_relu_kern<64, 8><<<dim3(16, 16), blk, 0, stream>>>(Bb, w3a, bn3a, A, 128, 128, 8, 8);
  adder_bn_relu_kern<128, 8><<<dim3(16, 16), blk, 0, stream>>>(A, w3b, bn3b, Bb, 128, 128, 8, 8);
  adder_bn_relu_kern<128, 8><<<dim3(16, 16), blk, 0, stream>>>(Bb, w3c, bn3c, A, 128, 128, 8, 8);
  adder_bn_relu_kern<128, 8><<<dim3(16, 16), blk, 0, stream>>>(A, w3d, bn3d, Bb, 128, 128, 8, 8);
  maxpool2_kern<<<dim3((128 * 128 * 4 * 4 + 255) / 256), blk, 0, stream>>>(Bb, A, 128, 8, 8, 128 * 128 * 4 * 4);

  // FC head: A is now [128, 2048] (NCHW flatten). fp32 WMMA GEMM.
  fc_wmma_kern<<<dim3(8), dim3(32), 0, stream>>>(A, fc_w, fc_b, out);
}